// TemporalPropagator_22385369547038
// MI455X (gfx1250) — compile-verified
//
#include <hip/hip_runtime.h>
#include <math.h>

// Problem dims (fixed by reference)
#define D      256
#define Bb     2
#define T      64
#define Nn     512
#define M_ROWS (Bb*T*Nn)   // 65536 rows for the GEMMs
#define NTD    (Nn*D)      // stride between t-steps in [B,T,N,D]

typedef float v2f __attribute__((ext_vector_type(2)));
typedef float v8f __attribute__((ext_vector_type(8)));

// ---- workspace layout (floats). Total = 393,472 floats ~ 1.54 MB ----
#define WS_QR_RE   0                       // Q column-major [e][d] (65536)
#define WS_QR_IM   (WS_QR_RE + D*D)
#define WS_PK_RE   (WS_QR_IM + D*D)       // WMMA B-fragment packed u (65536)
#define WS_PK_IM   (WS_PK_RE + D*D)
#define WS_DEC_RE  (WS_PK_IM + D*D)       // decay [B*T*D]
#define WS_DEC_IM  (WS_DEC_RE + Bb*T*D)
#define WS_FOR_RE  (WS_DEC_IM + Bb*T*D)   // forcing [B*T*D]
#define WS_FOR_IM  (WS_FOR_RE + Bb*T*D)
#define WS_NSCALE  (WS_FOR_IM + Bb*T*D)   // [D]

#define LDSW 260   // padded LDS row stride (floats): 260 % 64 = 4 -> conflict-free A frags

__device__ __forceinline__ v8f wmma_f32(v2f a, v2f b, v8f c) {
  // V_WMMA_F32_16X16X4_F32 : D = A(16x4 f32) * B(4x16 f32) + C(16x16 f32)
  return __builtin_amdgcn_wmma_f32_16x16x4_f32(false, a, false, b, (short)0, c,
                                               false, false);
}

// CDNA5 async memory->LDS copy (ASYNCcnt-tracked). lds_off = LDS byte address
// (low 32 bits of the generic shared pointer), gaddr = 64-bit global address.
__device__ __forceinline__ void async_ld_b128(unsigned lds_off, const void* gaddr) {
  asm volatile("global_load_async_to_lds_b128 %0, %1, off"
               :
               : "v"(lds_off), "v"((unsigned long long)(uintptr_t)gaddr)
               : "memory");
}
__device__ __forceinline__ void wait_async0() {
  asm volatile("s_wait_asynccnt 0x0" ::: "memory");
}

// -------------------------------------------------------------------------
// Stage 1: complex QR via modified Gram-Schmidt. One workgroup, 256 threads.
// Thread t owns row-element t of every column. Q stored column-major in ws.
// -------------------------------------------------------------------------
__global__ void qr_mgs_kernel(const float* __restrict__ Are,
                              const float* __restrict__ Aim,
                              float* __restrict__ ws) {
  float* qre = ws + WS_QR_RE;
  float* qim = ws + WS_QR_IM;
  const int t    = threadIdx.x;   // 0..255
  const int lane = t & 31;
  const int wv   = t >> 5;        // 0..7 (wave32)
  __shared__ float swr[8], swi[8];

  for (int j = 0; j < D; ++j) {
    float ar = Are[t * D + j];
    float ai = Aim[t * D + j];
    for (int i = 0; i < j; ++i) {
      const float qr = qre[i * D + t];   // own element, written by this thread
      const float qi = qim[i * D + t];
      // proj = conj(q_i) . a_j
      float pr = qr * ar + qi * ai;
      float pi = qr * ai - qi * ar;
#pragma unroll
      for (int off = 16; off > 0; off >>= 1) {
        pr += __shfl_xor(pr, off, 32);
        pi += __shfl_xor(pi, off, 32);
      }
      if (lane == 0) { swr[wv] = pr; swi[wv] = pi; }
      __syncthreads();
      float PR = 0.f, PI = 0.f;
#pragma unroll
      for (int w = 0; w < 8; ++w) { PR += swr[w]; PI += swi[w]; }
      __syncthreads();
      // a_j -= proj * q_i
      ar -= PR * qr - PI * qi;
      ai -= PR * qi + PI * qr;
    }
    float nn = ar * ar + ai * ai;
#pragma unroll
    for (int off = 16; off > 0; off >>= 1) nn += __shfl_xor(nn, off, 32);
    if (lane == 0) swr[wv] = nn;
    __syncthreads();
    float NN = 0.f;
#pragma unroll
    for (int w = 0; w < 8; ++w) NN += swr[w];
    __syncthreads();
    const float inv = rsqrtf(fmaxf(NN, 1e-30f));
    qre[j * D + t] = ar * inv;
    qim[j * D + t] = ai * inv;
  }
}

// -------------------------------------------------------------------------
// Stage 2: per-(b,t,d) decay & forcing; per-d noise scale.
// -------------------------------------------------------------------------
__global__ void coef_kernel(const float* __restrict__ dt,
                            const float* __restrict__ lam_re,
                            const float* __restrict__ lam_im,
                            const float* __restrict__ noise_raw,
                            float* __restrict__ ws) {
  const int idx = blockIdx.x * blockDim.x + threadIdx.x;  // B*T*D
  if (idx >= Bb * T * D) return;
  const int d  = idx & (D - 1);
  const int bt = idx >> 8;

  const float lr = lam_re[d];
  const float z  = -lr;
  const float sp = (z > 20.f) ? z : log1pf(expf(z));      // softplus(-lr)
  const float lam_r = fmaxf(-sp, -0.3f);                  // MAX_GROWTH_RATE
  const float lam_i = lam_im[d];

  const float dtv = dt[bt];
  const float er  = expf(dtv * lam_r);
  const float dre = er * cosf(dtv * lam_i);
  const float dim = er * sinf(dtv * lam_i);

  const float ssx = lam_r + 1e-12f;
  const float sg  = (ssx > 0.f) ? 1.f : ((ssx < 0.f) ? -1.f : 0.f);
  const float sr  = lam_r + 1e-8f * sg;
  const float si  = lam_i;
  const float nr  = dre - 1.f, ni = dim;
  const float d2  = sr * sr + si * si;
  ws[WS_DEC_RE + idx] = dre;
  ws[WS_DEC_IM + idx] = dim;
  ws[WS_FOR_RE + idx] = (nr * sr + ni * si) / d2;
  ws[WS_FOR_IM + idx] = (ni * sr - nr * si) / d2;

  if (bt == 0) {
    const float nz = noise_raw[d];
    const float nsp = (nz > 20.f) ? nz : log1pf(expf(nz));
    ws[WS_NSCALE + d] = nsp * 0.01f;   // INIT_NOISE_SCALE
  }
}

// -------------------------------------------------------------------------
// Stage 3: repack Q (column-major [e][k]) into WMMA B-fragment order:
// pack[((c*64 + s)*32 + lane)*2 + {0,1}] = u[k0][16c + (lane&15)], k0 = 4s + 2*(lane>>4)
// so each wave's B load is one coalesced 64-bit load per lane.
// -------------------------------------------------------------------------
__global__ void repack_kernel(float* __restrict__ ws) {
  const int idx = blockIdx.x * blockDim.x + threadIdx.x;  // 16*64*32
  if (idx >= 16 * 64 * 32) return;
  const int lane = idx & 31;
  const int s    = (idx >> 5) & 63;
  const int c    = idx >> 11;
  const int hi = lane >> 4, n = lane & 15;
  const int k0 = 4 * s + 2 * hi;
  const int e  = 16 * c + n;
  const float* qre = ws + WS_QR_RE;   // column-major: u[k][e] = q[e*D + k]
  const float* qim = ws + WS_QR_IM;
  float* pr = ws + WS_PK_RE;
  float* pi = ws + WS_PK_IM;
  const int o = ((c * 64 + s) * 32 + lane) * 2;
  pr[o]     = qre[e * D + k0];
  pr[o + 1] = qre[e * D + k0 + 1];
  pi[o]     = qim[e * D + k0];
  pi[o + 1] = qim[e * D + k0 + 1];
}

// -------------------------------------------------------------------------
// Stage 4: encode GEMM  x_enc = x @ conj(u), fused with forcing & noise.
// Writes b_term directly into d_out as [2,B,T,N,D] (re plane, im plane).
// WG = 128 threads (4 waves). Each wave owns TWO 16-row strips that share
// the B fragments -> 4 WMMA per {2 ds_load_b64 + 2 global_load_b64}.
// x tile staged to LDS with GLOBAL_LOAD_ASYNC_TO_LDS_B128 (ASYNCcnt).
// -------------------------------------------------------------------------
__global__ void encode_kernel(const float* __restrict__ x,
                              const float* __restrict__ dt,
                              const float* __restrict__ eps,
                              const float* __restrict__ ws,
                              float* __restrict__ out) {
  extern __shared__ float As[];                 // 128 * LDSW floats
  const int tid  = threadIdx.x;                 // 128
  const int lane = tid & 31;
  const int wv   = tid >> 5;                    // 4 waves, 32 rows each
  const int row0 = blockIdx.x * 128;
  const int bt   = row0 / Nn;                   // 128 | 512 -> constant per WG

  // async-stage x tile: 128 rows x 256 cols, 16B per lane per op
  for (int idx = tid; idx < 128 * 64; idx += 128) {
    const int r = idx >> 6, c4 = idx & 63;
    async_ld_b128((unsigned)(uintptr_t)(&As[r * LDSW + c4 * 4]),
                  x + (size_t)(row0 + r) * D + c4 * 4);
  }
  wait_async0();
  __syncthreads();

  const float* pkr = ws + WS_PK_RE;
  const float* pki = ws + WS_PK_IM;
  const int hi = lane >> 4, nl = lane & 15;
  const int m0 = wv * 32 + nl;                  // strip 0 A-frag row
  const int m1 = m0 + 16;                       // strip 1 A-frag row
  const float sq = sqrtf(fmaxf(dt[bt], 0.f));

  for (int c = 0; c < 16; ++c) {
    v8f ar0 = {0.f,0.f,0.f,0.f,0.f,0.f,0.f,0.f};
    v8f ai0 = {0.f,0.f,0.f,0.f,0.f,0.f,0.f,0.f};
    v8f ar1 = {0.f,0.f,0.f,0.f,0.f,0.f,0.f,0.f};
    v8f ai1 = {0.f,0.f,0.f,0.f,0.f,0.f,0.f,0.f};
    const float* pbr = pkr + ((size_t)(c * 64) * 32 + lane) * 2;
    const float* pbi = pki + ((size_t)(c * 64) * 32 + lane) * 2;
    for (int s = 0; s < 64; ++s) {
      const v2f a0 = *(const v2f*)(&As[m0 * LDSW + s * 4 + 2 * hi]);
      const v2f a1 = *(const v2f*)(&As[m1 * LDSW + s * 4 + 2 * hi]);
      const v2f br = *(const v2f*)(pbr + s * 64);
      const v2f bi = *(const v2f*)(pbi + s * 64);
      ar0 = wmma_f32(a0, br, ar0);
      ai0 = wmma_f32(a0, bi, ai0);
      ar1 = wmma_f32(a1, br, ar1);
      ai1 = wmma_f32(a1, bi, ai1);
    }
    const int e = c * 16 + nl;
    const float fre = ws[WS_FOR_RE + bt * D + e];
    const float fim = ws[WS_FOR_IM + bt * D + e];
    const float ns  = ws[WS_NSCALE + e] * sq;
#pragma unroll
    for (int i = 0; i < 8; ++i) {
      // strip 0
      {
        const int grow = row0 + wv * 32 + i + 8 * hi;  // C layout: VGPR i -> M=i / i+8
        const size_t off = (size_t)grow * D + e;
        const float xe_re = ar0[i];
        const float xe_im = -ai0[i];                   // x @ conj(u): -x@u_im
        const float ep = eps[off];
        out[off]                      = fre * xe_re - fim * xe_im + ep * ns;
        out[(size_t)M_ROWS * D + off] = fre * xe_im + fim * xe_re;  // eps is real
      }
      // strip 1
      {
        const int grow = row0 + wv * 32 + 16 + i + 8 * hi;
        const size_t off = (size_t)grow * D + e;
        const float xe_re = ar1[i];
        const float xe_im = -ai1[i];
        const float ep = eps[off];
        out[off]                      = fre * xe_re - fim * xe_im + ep * ns;
        out[(size_t)M_ROWS * D + off] = fre * xe_im + fim * xe_re;
      }
    }
  }
}

// -------------------------------------------------------------------------
// Stage 5: linear recurrence over T, in place in d_out.
// One thread per (b,n,d): h_t = decay_t * h_{t-1} + b_t, h_{-1} = 0.
// -------------------------------------------------------------------------
__global__ void scan_kernel(const float* __restrict__ ws, float* __restrict__ out) {
  const int tid = blockIdx.x * blockDim.x + threadIdx.x;   // B*N*D = 262144
  const int d = tid & (D - 1);
  const int n = (tid >> 8) & (Nn - 1);
  const int b = tid >> 17;                                 // / (N*D)
  const size_t base  = (size_t)b * T * NTD + (size_t)n * D + d;
  const int    cbase = b * T * D + d;
  float* ore = out;
  float* oim = out + (size_t)M_ROWS * D;
  float hr = 0.f, hi = 0.f;
  for (int t = 0; t < T; ++t) {
    const size_t idx = base + (size_t)t * NTD;
    if (t + 1 < T) {
      __builtin_prefetch(&ore[idx + NTD], 0, 1);   // global_prefetch_b8
      __builtin_prefetch(&oim[idx + NTD], 0, 1);
    }
    const float dr = ws[WS_DEC_RE + cbase + t * D];
    const float di = ws[WS_DEC_IM + cbase + t * D];
    const float br = ore[idx];
    const float bi = oim[idx];
    const float nhr = dr * hr - di * hi + br;
    const float nhi = dr * hi + di * hr + bi;
    hr = nhr; hi = nhi;
    ore[idx] = hr;
    oim[idx] = hi;
  }
}

// -------------------------------------------------------------------------
// Stage 6: decode GEMM  y = h @ u  (complex x complex), in place in d_out.
// WG = 128 threads (4 waves), 128-row tile; both h planes async-staged to
// LDS before any store so in-place overwrite is safe. Each wave owns TWO
// strips -> 8 WMMA per {4 ds_load_b64 + 2 global_load_b64}.
// -------------------------------------------------------------------------
__global__ void decode_kernel(const float* __restrict__ ws, float* __restrict__ out) {
  extern __shared__ float sh[];                 // 2 * 128 * LDSW floats (~266 KB)
  float* Ahr = sh;
  float* Ahi = sh + 128 * LDSW;
  const int tid  = threadIdx.x;                 // 128
  const int lane = tid & 31;
  const int wv   = tid >> 5;                    // 4 waves
  const int row0 = blockIdx.x * 128;
  const float* hre = out;
  const float* him = out + (size_t)M_ROWS * D;

  for (int idx = tid; idx < 128 * 64; idx += 128) {
    const int r = idx >> 6, c4 = idx & 63;
    async_ld_b128((unsigned)(uintptr_t)(&Ahr[r * LDSW + c4 * 4]),
                  hre + (size_t)(row0 + r) * D + c4 * 4);
    async_ld_b128((unsigned)(uintptr_t)(&Ahi[r * LDSW + c4 * 4]),
                  him + (size_t)(row0 + r) * D + c4 * 4);
  }
  wait_async0();
  __syncthreads();

  const float* pkr = ws + WS_PK_RE;
  const float* pki = ws + WS_PK_IM;
  const int hi = lane >> 4, nl = lane & 15;
  const int m0 = wv * 32 + nl;
  const int m1 = m0 + 16;

  for (int c = 0; c < 16; ++c) {
    v8f rr0 = {0.f,0.f,0.f,0.f,0.f,0.f,0.f,0.f};
    v8f ii0 = {0.f,0.f,0.f,0.f,0.f,0.f,0.f,0.f};
    v8f ri0 = {0.f,0.f,0.f,0.f,0.f,0.f,0.f,0.f};
    v8f ir0 = {0.f,0.f,0.f,0.f,0.f,0.f,0.f,0.f};
    v8f rr1 = {0.f,0.f,0.f,0.f,0.f,0.f,0.f,0.f};
    v8f ii1 = {0.f,0.f,0.f,0.f,0.f,0.f,0.f,0.f};
    v8f ri1 = {0.f,0.f,0.f,0.f,0.f,0.f,0.f,0.f};
    v8f ir1 = {0.f,0.f,0.f,0.f,0.f,0.f,0.f,0.f};
    const float* pbr = pkr + ((size_t)(c * 64) * 32 + lane) * 2;
    const float* pbi = pki + ((size_t)(c * 64) * 32 + lane) * 2;
    for (int s = 0; s < 64; ++s) {
      const v2f a_r0 = *(const v2f*)(&Ahr[m0 * LDSW + s * 4 + 2 * hi]);
      const v2f a_i0 = *(const v2f*)(&Ahi[m0 * LDSW + s * 4 + 2 * hi]);
      const v2f a_r1 = *(const v2f*)(&Ahr[m1 * LDSW + s * 4 + 2 * hi]);
      const v2f a_i1 = *(const v2f*)(&Ahi[m1 * LDSW + s * 4 + 2 * hi]);
      const v2f br = *(const v2f*)(pbr + s * 64);
      const v2f bi = *(const v2f*)(pbi + s * 64);
      rr0 = wmma_f32(a_r0, br, rr0);
      ii0 = wmma_f32(a_i0, bi, ii0);
      ri0 = wmma_f32(a_r0, bi, ri0);
      ir0 = wmma_f32(a_i0, br, ir0);
      rr1 = wmma_f32(a_r1, br, rr1);
      ii1 = wmma_f32(a_i1, bi, ii1);
      ri1 = wmma_f32(a_r1, bi, ri1);
      ir1 = wmma_f32(a_i1, br, ir1);
    }
    const int e = c * 16 + nl;
#pragma unroll
    for (int i = 0; i < 8; ++i) {
      {
        const int grow = row0 + wv * 32 + i + 8 * hi;
        const size_t off = (size_t)grow * D + e;
        out[off]                      = rr0[i] - ii0[i];   // y_re
        out[(size_t)M_ROWS * D + off] = ri0[i] + ir0[i];   // y_im
      }
      {
        const int grow = row0 + wv * 32 + 16 + i + 8 * hi;
        const size_t off = (size_t)grow * D + e;
        out[off]                      = rr1[i] - ii1[i];
        out[(size_t)M_ROWS * D + off] = ri1[i] + ir1[i];
      }
    }
  }
}

// -------------------------------------------------------------------------
extern "C" void kernel_launch(void* const* d_in, const int* in_sizes, int n_in,
                              void* d_out, int out_size, void* d_ws, size_t ws_size,
                              hipStream_t stream) {
  const float* x    = (const float*)d_in[0];
  const float* dt   = (const float*)d_in[1];
  const float* eps  = (const float*)d_in[2];
  const float* ure  = (const float*)d_in[3];
  const float* uim  = (const float*)d_in[4];
  const float* lre  = (const float*)d_in[5];
  const float* lim  = (const float*)d_in[6];
  const float* nraw = (const float*)d_in[7];
  float* out = (float*)d_out;
  float* ws  = (float*)d_ws;   // needs ~1.54 MB

  qr_mgs_kernel<<<1, 256, 0, stream>>>(ure, uim, ws);
  coef_kernel<<<(Bb * T * D) / 256, 256, 0, stream>>>(dt, lre, lim, nraw, ws);
  repack_kernel<<<(16 * 64 * 32) / 256, 256, 0, stream>>>(ws);

  const size_t ldsE = (size_t)128 * LDSW * sizeof(float);      // 133,120 B
  encode_kernel<<<M_ROWS / 128, 128, ldsE, stream>>>(x, dt, eps, ws, out);

  scan_kernel<<<(Bb * Nn * D) / 256, 256, 0, stream>>>(ws, out);

  const size_t ldsD = (size_t)2 * 128 * LDSW * sizeof(float);  // 266,240 B
  decode_kernel<<<M_ROWS / 128, 128, ldsD, stream>>>(ws, out);
}